// fTNModel_41936060678471
// MI455X (gfx1250) — compile-verified
//
#include <hip/hip_runtime.h>

typedef float v2f __attribute__((ext_vector_type(2)));
typedef float v8f __attribute__((ext_vector_type(8)));

#define LX 8
#define LY 8
#define DBOND 5
#define BCFG 8
// max state per config: K*N = 25 * 5^7 = 1953125 floats
#define BUFELE 1953125

constexpr int ipow(int b, int e) { return e <= 0 ? 1 : b * ipow(b, e - 1); }

// Compile-time shape bundle for one site. XRC: 0=first row, 1=mid rows, 2=last row.
template <int XRC, int Y>
struct SiteDims {
    static constexpr int  DU     = (XRC < 2) ? DBOND : 1;
    static constexpr int  DD     = (XRC > 0) ? DBOND : 1;
    static constexpr int  DL     = (Y > 0) ? DBOND : 1;
    static constexpr int  DR     = (Y < LY - 1) ? DBOND : 1;
    static constexpr int  A      = ipow(DU, Y);             // a-count entering site
    static constexpr int  R      = ipow(DD, LY - 1 - Y);    // remaining down bonds
    static constexpr int  RN     = (Y == LY - 1) ? 1 : ipow(DD, LY - 2 - Y);
    static constexpr int  K      = DL * DD;
    static constexpr int  M      = DU * DR;
    static constexpr int  N      = A * R;
    static constexpr int  NNEXT  = A * DU * RN;
    static constexpr int  KSTEPS = (K + 3) / 4;
    static constexpr bool FIRST  = (XRC == 0 && Y == 0);
};

// One site absorption: C[m=(u,r)][n=(a,rr)] = sum_k T2[m][k] * S[k][n], k=(l,d),
// scatter-stored (a pure index-digit permutation) into the next site's K-major layout.
// Each wave: 2 M-tiles x 2 N-tiles (A fragments reused across both B tiles).
template <int XRC, int Y>
__global__ __launch_bounds__(128) void peps_site_step(
    const float* __restrict__ peps, const int* __restrict__ xcfg,
    const float* __restrict__ src, float* __restrict__ dst, int xr)
{
    using SD = SiteDims<XRC, Y>;

    __shared__ float Tt[32][29];   // zero-padded T2[m][k]; stride 29: conflict-free

    const int b   = blockIdx.y;
    const int tid = threadIdx.x;
    const int phys = xcfg[b * (LX * LY) + xr * LY + Y];
    // peps strides (elements): xr:10000 y:1250 u:250 r:50 d:10 l:2 phys:1
    const float* pbase = peps + xr * 10000 + Y * 1250 + phys;

    #pragma unroll
    for (int idx = tid; idx < 32 * 28; idx += 128) {
        const int m = idx / 28, k = idx % 28;
        float v = 0.0f;
        if (m < SD::M && k < SD::K) {
            v = pbase[(m / SD::DR) * 250 + (m % SD::DR) * 50 +
                      (k % SD::DD) * 10 + (k / SD::DD) * 2];
        }
        Tt[m][k] = v;
    }
    __syncthreads();

    const int wave = tid >> 5;
    const int lane = tid & 31;
    const int lh   = lane >> 4;      // 0: lanes 0-15 (K pair +0), 1: lanes 16-31 (K pair +2)
    const int ln   = lane & 15;
    const int n0   = (blockIdx.x * 4 + wave) * 32;   // two 16-col tiles per wave
    if (n0 >= SD::N) return;         // wave-uniform

    const float* sb = src + b * BUFELE;
    float*       db = dst + b * BUFELE;
    const int nA  = n0 + ln;
    const int nB  = n0 + 16 + ln;
    const int ncA = (nA < SD::N) ? nA : (SD::N - 1);   // clamped: loads in-bounds
    const int ncB = (nB < SD::N) ? nB : (SD::N - 1);
    const bool colA = (nA < SD::N);
    const bool colB = (nB < SD::N);

    v8f c00 = {}, c10 = {}, c01 = {}, c11 = {};
    #pragma unroll
    for (int ks = 0; ks < SD::KSTEPS; ++ks) {
        const int k = ks * 4 + 2 * lh;
        v2f a0, a1, bA, bB;
        a0.x = Tt[ln][k];        a0.y = Tt[ln][k + 1];
        a1.x = Tt[16 + ln][k];   a1.y = Tt[16 + ln][k + 1];
        if constexpr (SD::FIRST) {   // implicit state = [1.0] (K=N=1)
            bA.x = (k == 0 && nA == 0) ? 1.0f : 0.0f;  bA.y = 0.0f;
            bB.x = 0.0f;                               bB.y = 0.0f;
        } else {
            const int k0 = (k     < SD::K) ? k       : (SD::K - 1);
            const int k1 = (k + 1 < SD::K) ? (k + 1) : (SD::K - 1);
            const float lA0 = sb[k0 * SD::N + ncA];
            const float lA1 = sb[k1 * SD::N + ncA];
            const float lB0 = sb[k0 * SD::N + ncB];
            const float lB1 = sb[k1 * SD::N + ncB];
            bA.x = (colA && (k     < SD::K)) ? lA0 : 0.0f;
            bA.y = (colA && (k + 1 < SD::K)) ? lA1 : 0.0f;
            bB.x = (colB && (k     < SD::K)) ? lB0 : 0.0f;
            bB.y = (colB && (k + 1 < SD::K)) ? lB1 : 0.0f;
        }
        c00 = __builtin_amdgcn_wmma_f32_16x16x4_f32(false, a0, false, bA,
                                                    (short)0, c00, false, false);
        c10 = __builtin_amdgcn_wmma_f32_16x16x4_f32(false, a1, false, bA,
                                                    (short)0, c10, false, false);
        c01 = __builtin_amdgcn_wmma_f32_16x16x4_f32(false, a0, false, bB,
                                                    (short)0, c01, false, false);
        c11 = __builtin_amdgcn_wmma_f32_16x16x4_f32(false, a1, false, bB,
                                                    (short)0, c11, false, false);
    }

    // Scatter one 16-col tile: dest = (r*DD+d2)*NNEXT + (a*DU+u)*RN + rr2,
    // u=m/DR, r=m%DR. Per-lane base hoisted; per-element part is a select
    // between two compile-time offsets (m differs only via lane-half).
    auto scatter = [&](int n, bool ncol, const v8f& cc0, const v8f& cc1) {
        const int a    = n / SD::R;     // constant power-of-5 divisors
        const int rr   = n % SD::R;
        const int d2   = rr / SD::RN;
        const int rr2  = rr % SD::RN;
        const int base = d2 * SD::NNEXT + (a * SD::DU) * SD::RN + rr2;
        #pragma unroll
        for (int t = 0; t < 2; ++t) {
            const v8f& cc = t ? cc1 : cc0;
            #pragma unroll
            for (int vg = 0; vg < 8; ++vg) {
                const int m0 = t * 16 + vg;      // literal after unroll
                const int m1 = m0 + 8;
                const int off0 = (m0 % SD::DR) * (SD::DD * SD::NNEXT) + (m0 / SD::DR) * SD::RN;
                const int off1 = (m1 % SD::DR) * (SD::DD * SD::NNEXT) + (m1 / SD::DR) * SD::RN;
                const bool v0 = (m0 < SD::M), v1 = (m1 < SD::M);   // fold to constants
                if (v0 || v1) {
                    const bool okm = lh ? v1 : v0;
                    const int  off = lh ? off1 : off0;
                    if (okm && ncol) db[base + off] = cc[vg];
                }
            }
        }
    };
    scatter(nA, colA, c00, c10);
    scatter(nB, colB, c01, c11);
}

__global__ void peps_gather(const float* __restrict__ fin, float* __restrict__ out)
{
    const int b = threadIdx.x;
    if (b < BCFG) out[b] = fin[b * BUFELE];
}

template <int XRC, int Y>
static inline void run_site(const float* peps, const int* x, float* h0, float* h1,
                            int& p, int xr, hipStream_t stream)
{
    using SD = SiteDims<XRC, Y>;
    float* src = p ? h1 : h0;
    float* dst = p ? h0 : h1;
    dim3 grid((unsigned)((SD::N + 127) / 128), BCFG, 1);
    peps_site_step<XRC, Y><<<grid, 128, 0, stream>>>(peps, x, src, dst, xr);
    p ^= 1;
}

template <int XRC>
static inline void run_row(const float* peps, const int* x, float* h0, float* h1,
                           int& p, int xr, hipStream_t stream)
{
    run_site<XRC, 0>(peps, x, h0, h1, p, xr, stream);
    run_site<XRC, 1>(peps, x, h0, h1, p, xr, stream);
    run_site<XRC, 2>(peps, x, h0, h1, p, xr, stream);
    run_site<XRC, 3>(peps, x, h0, h1, p, xr, stream);
    run_site<XRC, 4>(peps, x, h0, h1, p, xr, stream);
    run_site<XRC, 5>(peps, x, h0, h1, p, xr, stream);
    run_site<XRC, 6>(peps, x, h0, h1, p, xr, stream);
    run_site<XRC, 7>(peps, x, h0, h1, p, xr, stream);
}

extern "C" void kernel_launch(void* const* d_in, const int* in_sizes, int n_in,
                              void* d_out, int out_size, void* d_ws, size_t ws_size,
                              hipStream_t stream)
{
    const int*   x    = (const int*)d_in[0];    // (B, 64) int32 configs
    const float* peps = (const float*)d_in[1];  // (8,8,5,5,5,5,2) f32

    float* half0 = (float*)d_ws;
    float* half1 = half0 + (size_t)BCFG * BUFELE;

    int p = 0;   // ping-pong parity
    run_row<0>(peps, x, half0, half1, p, 0, stream);
    for (int xr = 1; xr < LX - 1; ++xr)
        run_row<1>(peps, x, half0, half1, p, xr, stream);
    run_row<2>(peps, x, half0, half1, p, LX - 1, stream);

    // 64 steps -> p back to 0; final dst was half0
    float* fin = p ? half1 : half0;
    peps_gather<<<1, 32, 0, stream>>>(fin, (float*)d_out);
}